// CannyEdge_81587198755157
// MI455X (gfx1250) — compile-verified
//
#include <hip/hip_runtime.h>
#include <stdint.h>

typedef __attribute__((ext_vector_type(2))) float v2f;
typedef __attribute__((ext_vector_type(8))) float v8f;

#define TS 48          // output tile (48x48 mag/idx per workgroup)
#define RS 54          // staged input region / LDS row stride (floats)
#define IMG_HW 262144  // 512*512
#define W512 512

// ---- Gaussian taps (ksize=5, sigma=1.4), normalized; branchless band ----
__device__ __forceinline__ float gband(int d) {
    float r = 0.0f;
    r = ((d == 0) | (d == 4)) ? 0.11020947f : r;
    r = ((d == 1) | (d == 3)) ? 0.23691122f : r;
    r = (d == 2) ? 0.30575730f : r;
    return r;
}

// ---- CDNA5 async copy: global -> LDS (tracked by ASYNCcnt) ----
__device__ __forceinline__ void async_load_f32(unsigned lds_addr, const float* g) {
    unsigned long long ga = (unsigned long long)(uintptr_t)g;
    asm volatile("global_load_async_to_lds_b32 %0, %1, off"
                 :: "v"(lds_addr), "v"(ga)
                 : "memory");
}
__device__ __forceinline__ void wait_async0() {
    asm volatile("s_wait_asynccnt 0" ::: "memory");
}

// =====================================================================
// Pass A: fused channel-sum + 5x5 reflect blur (WMMA) + Sobel + mag/idx
// grid = (11, 11, 16), block = 256
// =====================================================================
__global__ __launch_bounds__(256) void canny_blur_mag(
        const float* __restrict__ img,
        float* __restrict__ magbuf,
        unsigned char* __restrict__ idxbuf) {
    __shared__ float sIn0[RS * RS];   // ch0, then channel-sum S
    __shared__ float sIn1[RS * RS];   // ch1, then H-blurred Bh
    __shared__ float sIn2[RS * RS];   // ch2, then blurred Bv

    float* S  = sIn0;
    float* Bh = sIn1;
    float* Bv = sIn2;

    const int tid  = threadIdx.x;
    const int lane = tid & 31;
    const int wave = tid >> 5;
    const int b    = blockIdx.z;
    const int R0   = blockIdx.y * TS;
    const int C0   = blockIdx.x * TS;
    const int gr0  = R0 - 3, gc0 = C0 - 3;

    // ---- stage 3 channels into LDS with async copies, reflect-mapped ----
    const float* base = img + (size_t)b * 3u * IMG_HW;
    for (int e = tid; e < RS * RS; e += 256) {
        int li = e / RS, lj = e - li * RS;
        int r = gr0 + li; r = (r < 0) ? -r : ((r > 511) ? 1022 - r : r);
        int c = gc0 + lj; c = (c < 0) ? -c : ((c > 511) ? 1022 - c : c);
        int off = (r << 9) + c;
        async_load_f32((unsigned)(uintptr_t)&sIn0[e], base + off);
        async_load_f32((unsigned)(uintptr_t)&sIn1[e], base + IMG_HW + off);
        async_load_f32((unsigned)(uintptr_t)&sIn2[e], base + 2 * IMG_HW + off);
    }
    wait_async0();
    __syncthreads();

    // ---- channel sum (conv is linear; all ref output channels identical) ----
    for (int e = tid; e < RS * RS; e += 256)
        sIn0[e] = sIn0[e] + sIn1[e] + sIn2[e];
    __syncthreads();

    const int m    = lane & 15;   // M (A row / D col) index
    const int half = lane >> 4;   // K-half select per ISA A/B layout

    // ---- precompute banded coefficient fragments (per-lane constants) ----
    // bf[c5] holds {g[k0-m], g[k0+1-m]} with k0 = 4*c5 + 2*half.
    // Used as the B fragment in the H pass and the A fragment in the V pass.
    v2f bf[5];
    #pragma unroll
    for (int c5 = 0; c5 < 5; ++c5) {
        int k0 = c5 * 4 + half * 2;
        bf[c5].x = gband(k0 - m);
        bf[c5].y = gband(k0 + 1 - m);
    }

    // ---- Horizontal blur via WMMA: Bh[li][wc] = sum_k S[li][wc+k]*g[k] ----
    // Out(16x16) = In(16x20) x Band(20x16), 5 chained K=4 f32 WMMAs.
    {
        const int rtab[4] = {0, 16, 32, 38};   // row tiles cover li 0..53
        const int ctab[4] = {0, 16, 32, 34};   // col tiles cover wc 0..49
        #pragma unroll
        for (int tt = 0; tt < 2; ++tt) {
            int t  = wave + tt * 8;            // 16 tiles / 8 waves
            int rt = rtab[t >> 2], ct = ctab[t & 3];
            const float* arow = &S[(rt + m) * RS + ct + half * 2];
            v8f acc = {0.f, 0.f, 0.f, 0.f, 0.f, 0.f, 0.f, 0.f};
            #pragma unroll
            for (int c5 = 0; c5 < 5; ++c5) {
                v2f a;
                a.x = arow[c5 * 4];
                a.y = arow[c5 * 4 + 1];
                acc = __builtin_amdgcn_wmma_f32_16x16x4_f32(
                          false, a, false, bf[c5], (short)0, acc, false, false);
            }
            #pragma unroll
            for (int i = 0; i < 8; ++i)        // D: VGPR i -> row i (+8 for hi lanes)
                Bh[(rt + half * 8 + i) * RS + ct + m] = acc[i];
        }
    }
    __syncthreads();

    // ---- Vertical blur via WMMA: Bv[vr][wc] = sum_k g[k]*Bh[vr+k][wc] ----
    {
        const int vtab[4] = {0, 16, 32, 34};   // rows cover vr 0..49
        const int ctab[4] = {0, 16, 32, 34};
        #pragma unroll
        for (int tt = 0; tt < 2; ++tt) {
            int t  = wave + tt * 8;
            int vt = vtab[t >> 2], ct = ctab[t & 3];
            const float* bcol = &Bh[(vt + half * 2) * RS + ct + m];
            v8f acc = {0.f, 0.f, 0.f, 0.f, 0.f, 0.f, 0.f, 0.f};
            #pragma unroll
            for (int c5 = 0; c5 < 5; ++c5) {
                v2f bb;
                bb.x = bcol[(c5 * 4) * RS];
                bb.y = bcol[(c5 * 4 + 1) * RS];
                acc = __builtin_amdgcn_wmma_f32_16x16x4_f32(
                          false, bf[c5], false, bb, (short)0, acc, false, false);
            }
            #pragma unroll
            for (int i = 0; i < 8; ++i)
                Bv[(vt + half * 8 + i) * RS + ct + m] = acc[i];
        }
    }
    __syncthreads();

    // ---- Sobel (zero pad at image border) + mag + direction bucket ----
    #pragma unroll
    for (int p = 0; p < 9; ++p) {              // 48*48 = 2304 = 9*256
        int pix = tid + p * 256;
        int y = pix / TS, x = pix - y * TS;
        int gy = R0 + y, gx = C0 + x;
        int vr = y + 1, wc = x + 1;            // Bv[0][0] <-> (R0-1, C0-1)
        float bz[3][3];
        #pragma unroll
        for (int dy = -1; dy <= 1; ++dy) {
            #pragma unroll
            for (int dx = -1; dx <= 1; ++dx) {
                int yy = gy + dy, xx = gx + dx;
                bool inr = (yy >= 0) & (yy < W512) & (xx >= 0) & (xx < W512);
                float v = Bv[(vr + dy) * RS + (wc + dx)];
                bz[dy + 1][dx + 1] = inr ? v : 0.0f;
            }
        }
        float gxv = (bz[0][0] - bz[0][2]) + 2.0f * (bz[1][0] - bz[1][2]) +
                    (bz[2][0] - bz[2][2]);
        float gyv = (bz[0][0] + 2.0f * bz[0][1] + bz[0][2]) -
                    (bz[2][0] + 2.0f * bz[2][1] + bz[2][2]);
        float mg = sqrtf(gxv * gxv + gyv * gyv);
        float th = atan2f(gyv, gxv) * 57.29577951308232f + 180.0f;
        int id = ((int)floorf(th * 0.02222222222f)) & 7;   // floor(th/45) % 8
        if ((gy < W512) & (gx < W512)) {
            int o = (b << 18) + (gy << 9) + gx;
            magbuf[o] = mg;
            idxbuf[o] = (unsigned char)id;
        }
    }
}

// =====================================================================
// Pass B: dmin[d][h][w] = min over 16 batches of 3*(mag - shifted mag)
// (reference's jnp.min over concatenated batch axis). grid = 1024 x 256
// =====================================================================
__global__ __launch_bounds__(256) void canny_dmin(
        const float* __restrict__ magbuf, float* __restrict__ dminbuf) {
    int pix = blockIdx.x * 256 + threadIdx.x;          // 0 .. 262143
    int h = pix >> 9, w = pix & 511;
    const int dr[8] = {0, -1, -1, -1, 0, 1, 1, 1};
    const int dc[8] = {1, 1, 0, -1, -1, -1, 0, 1};
    float md[8];
    #pragma unroll
    for (int d = 0; d < 8; ++d) md[d] = 3.0e38f;
    for (int b = 0; b < 16; ++b) {
        int base = b << 18;
        float c = magbuf[base + pix];
        #pragma unroll
        for (int d = 0; d < 8; ++d) {
            int hh = h + dr[d], ww = w + dc[d];
            bool inr = (hh >= 0) & (hh < W512) & (ww >= 0) & (ww < W512);
            float n = inr ? magbuf[base + (hh << 9) + ww] : 0.0f;
            md[d] = fminf(md[d], 3.0f * (c - n));
        }
    }
    #pragma unroll
    for (int d = 0; d < 8; ++d) dminbuf[(d << 18) + pix] = md[d];
}

// =====================================================================
// Pass C: apply 4 orientation masks, replicate to 3 channels.
// grid = 16384 x 256
// =====================================================================
__global__ __launch_bounds__(256) void canny_nms(
        const float* __restrict__ magbuf, const unsigned char* __restrict__ idxbuf,
        const float* __restrict__ dminbuf, float* __restrict__ out) {
    int i = blockIdx.x * 256 + threadIdx.x;            // 0 .. 16*262144-1
    int b = i >> 18;
    int pix = i & (IMG_HW - 1);
    float mg = magbuf[i];
    int id = (int)idxbuf[i];
    float f = 1.0f;
    #pragma unroll
    for (int ni = 0; ni < 4; ++ni) {
        int pi = ni + 4;
        float dm = fminf(dminbuf[(pi << 18) + pix], dminbuf[(ni << 18) + pix]);
        bool is_max = dm > 0.0f;
        bool orient = (id == pi) | (id == ni);
        f = ((!is_max) & orient) ? 0.0f : f;
    }
    float r = mg * f;
    int ob = (b * 3) << 18;
    out[ob + pix] = r;
    out[ob + IMG_HW + pix] = r;
    out[ob + 2 * IMG_HW + pix] = r;
}

// =====================================================================
extern "C" void kernel_launch(void* const* d_in, const int* in_sizes, int n_in,
                              void* d_out, int out_size, void* d_ws, size_t ws_size,
                              hipStream_t stream) {
    (void)in_sizes; (void)n_in; (void)out_size; (void)ws_size;
    const float* img = (const float*)d_in[0];
    float* out = (float*)d_out;

    // workspace layout: mag (16 MiB f32) | idx (4 MiB u8) | dmin (8 MiB f32)
    float* magbuf = (float*)d_ws;
    unsigned char* idxbuf = (unsigned char*)((char*)d_ws + (size_t)16 * IMG_HW * 4);
    float* dminbuf = (float*)((char*)d_ws + (size_t)16 * IMG_HW * 4 + (size_t)16 * IMG_HW);

    dim3 gA(11, 11, 16);
    canny_blur_mag<<<gA, 256, 0, stream>>>(img, magbuf, idxbuf);
    canny_dmin<<<1024, 256, 0, stream>>>(magbuf, dminbuf);
    canny_nms<<<16384, 256, 0, stream>>>(magbuf, idxbuf, dminbuf, out);
}